// Loss_76063870812616
// MI455X (gfx1250) — compile-verified
//
#include <hip/hip_runtime.h>
#include <hip/hip_bf16.h>

// Problem constants (B=1, C=3, H=1080, W=1920 from the reference setup).
#define HH   1080
#define WW   1920
#define NPIX (HH * WW)          // 2,073,600 -> exactly 8100 blocks of 256
#define CC   3
#define NTOT (CC * NPIX)        // 6,220,800 elements averaged
#define BLK  256
#define NBLK (NPIX / BLK)       // 8100

typedef __attribute__((ext_vector_type(2))) float v2f;
typedef __attribute__((ext_vector_type(8))) float v8f;

// ---------------------------------------------------------------------------
// Wave-wide (32-lane) f32 sum using V_WMMA_F32_16X16X4_F32.
// A (16x4): lane m (m<16) holds {v_m, 0} (K=0,1); lane m+16 holds {v_{m+16},0}
// (K=2,3).  B = all-ones (4x16).  D[m,n] = rowsum_m = v_m + v_{m+16}.
// Per-lane sum of the 8 D VGPRs gives sum over M=0..7 (lanes 0-15) or
// M=8..15 (lanes 16-31); adding across the half-wave boundary yields the
// full 32-lane total in every lane.  Exact f32 arithmetic (RNE).
// EXEC must be all ones at the call site (full 256-thread blocks, no early
// returns before the reduction).
// ---------------------------------------------------------------------------
__device__ __forceinline__ float wave_sum_wmma(float v) {
  v2f a; a.x = v;    a.y = 0.0f;
  v2f b; b.x = 1.0f; b.y = 1.0f;     // all-ones B regardless of lane layout
  v8f c = {};
  c = __builtin_amdgcn_wmma_f32_16x16x4_f32(
      /*neg_a=*/false, a, /*neg_b=*/false, b,
      /*c_mod=*/(short)0, c, /*reuse_a=*/false, /*reuse_b=*/false);
  float s = c[0] + c[1] + c[2] + c[3] + c[4] + c[5] + c[6] + c[7];
  s += __shfl_xor(s, 16, 32);        // combine the two half-wave sums
  return s;                          // total in every lane
}

// Block (256-thread) reduction: WMMA wave sums -> LDS -> wave 0 combines.
// Returns the block total in wave 0 (valid at threadIdx.x == 0).
__device__ __forceinline__ float block_sum(float v) {
  __shared__ float sh[BLK / 32];
  const int lane = threadIdx.x & 31;
  const int wid  = threadIdx.x >> 5;
  float w = wave_sum_wmma(v);
  if (lane == 0) sh[wid] = w;
  __syncthreads();
  float r = 0.0f;
  if (wid == 0) {
    float t = (lane < (BLK / 32)) ? sh[lane] : 0.0f;
    t += __shfl_xor(t, 1, 32);
    t += __shfl_xor(t, 2, 32);
    t += __shfl_xor(t, 4, 32);
    r = t;
  }
  return r;
}

// torch bicubic weights, a = -0.75, taps at offsets -1,0,1,2
__device__ __forceinline__ void cubic_weights(float t, float w[4]) {
  const float A = -0.75f;
  float t1 = t + 1.0f;
  w[0] = ((A * t1 - 5.0f * A) * t1 + 8.0f * A) * t1 - 4.0f * A;
  w[1] = ((A + 2.0f) * t - (A + 3.0f)) * t * t + 1.0f;
  float u = 1.0f - t;
  w[2] = ((A + 2.0f) * u - (A + 3.0f)) * u * u + 1.0f;
  w[3] = 1.0f - w[0] - w[1] - w[2];
}

// ---------------------------------------------------------------------------
// Pass 1: occlusion map from flow finite differences (u8, 2 MB in d_ws).
//   a[y,x] = flow_x[y+1,x]-flow_x[y,x]  (y < H-1, else 0)
//   b[y,x] = flow_y[y,x+1]-flow_y[y,x]  (x < W-1, else 0)
//   occ    = |a+b| > 0.75
// ---------------------------------------------------------------------------
__global__ __launch_bounds__(BLK)
void occ_kernel(const float* __restrict__ flow, const int* __restrict__ flag,
                unsigned char* __restrict__ occ) {
  if (flag[0]) return;                       // uniform: no_warping path
  const int pix = blockIdx.x * BLK + threadIdx.x;
  const int y = pix / WW;
  const int x = pix - y * WW;
  float a = 0.0f, b = 0.0f;
  if (y < HH - 1) a = flow[pix + WW] - flow[pix];
  if (x < WW - 1) b = flow[NPIX + pix + 1] - flow[NPIX + pix];
  occ[pix] = (fabsf(a + b) > 0.75f) ? (unsigned char)1 : (unsigned char)0;
}

// ---------------------------------------------------------------------------
// Pass 2 (fused): dilation + in-range mask + bicubic warp + masked L1,
// block-reduced into partials[blockIdx.x].
// ---------------------------------------------------------------------------
__global__ __launch_bounds__(BLK)
void loss_kernel(const float* __restrict__ in1,  const float* __restrict__ prev,
                 const float* __restrict__ flow, const float* __restrict__ m0,
                 const float* __restrict__ ex,   const int*   __restrict__ flag,
                 const unsigned char* __restrict__ occ,
                 float* __restrict__ partials) {
  const int pix = blockIdx.x * BLK + threadIdx.x;
  float sum = 0.0f;

  if (flag[0]) {
    // no_warping: mean |input1 - prev1|
    sum  = fabsf(in1[pix]            - prev[pix]);
    sum += fabsf(in1[NPIX + pix]     - prev[NPIX + pix]);
    sum += fabsf(in1[2 * NPIX + pix] - prev[2 * NPIX + pix]);
  } else {
    // kick off streaming of the pointwise operands (global_prefetch_b8)
    __builtin_prefetch(prev + pix, 0, 3);
    __builtin_prefetch(m0 + pix, 0, 3);
    __builtin_prefetch(ex + pix, 0, 3);

    const int y = pix / WW;
    const int x = pix - y * WW;
    const float fx = flow[pix];
    const float fy = flow[NPIX + pix];
    const float gx = 2.0f * ((float)x + fx) / (float)(WW - 1) - 1.0f;
    const float gy = 2.0f * ((float)y + fy) / (float)(HH - 1) - 1.0f;
    const float inr = (gx >= -1.0f && gx <= 1.0f &&
                       gy >= -1.0f && gy <= 1.0f) ? 1.0f : 0.0f;

    // 4x4 binary dilation over occ, window rows y-1..y+2 / cols x-1..x+2;
    // 2-pixel border forced occluded (reference bicubic branch).
    float dil;
    if (y < 2 || y >= HH - 2 || x < 2 || x >= WW - 2) {
      dil = 1.0f;
    } else {
      unsigned int d = 0;
      #pragma unroll
      for (int dy = -1; dy <= 2; ++dy) {
        const unsigned char* r = occ + (y + dy) * WW + x;   // rows in [1,H-1]
        d |= (unsigned int)r[-1] | r[0] | r[1] | r[2];      // cols in [1,W-1]
      }
      dil = d ? 1.0f : 0.0f;
    }

    const float mp = inr * (1.0f - dil);
    if (mp != 0.0f) {
      // bicubic gather (border clamp), identical index math to the reference
      const float ixf = (gx + 1.0f) * 0.5f * (float)(WW - 1);
      const float iyf = (gy + 1.0f) * 0.5f * (float)(HH - 1);
      const float x0 = floorf(ixf);
      const float y0 = floorf(iyf);
      float wx[4], wy[4];
      cubic_weights(ixf - x0, wx);
      cubic_weights(iyf - y0, wy);
      const int x0i = (int)x0, y0i = (int)y0;
      int xi[4], yi[4];
      #pragma unroll
      for (int j = 0; j < 4; ++j) {
        int v = x0i + j - 1;
        xi[j] = min(max(v, 0), WW - 1);
      }
      #pragma unroll
      for (int i = 0; i < 4; ++i) {
        int v = y0i + i - 1;
        yi[i] = min(max(v, 0), HH - 1);
      }
      #pragma unroll
      for (int c = 0; c < CC; ++c) {
        const float* im = in1 + c * NPIX;
        float acc = 0.0f;
        #pragma unroll
        for (int i = 0; i < 4; ++i) {
          const float* rp = im + yi[i] * WW;
          float rv = wx[0] * rp[xi[0]] + wx[1] * rp[xi[1]] +
                     wx[2] * rp[xi[2]] + wx[3] * rp[xi[3]];
          acc += wy[i] * rv;
        }
        const float M = m0[c * NPIX + pix] * mp * ex[c * NPIX + pix];
        sum += fabsf(M * acc - M * prev[c * NPIX + pix]);
      }
    }
    // mp == 0 => Mask1 == 0 => |0 - 0| = 0 for all channels: skip gathers.
  }

  // all 256 threads reach here (full EXEC for the WMMA reduction)
  float bs = block_sum(sum);
  if (threadIdx.x == 0) partials[blockIdx.x] = bs;
}

// ---------------------------------------------------------------------------
// Pass 3: deterministic final reduction of 8100 block partials -> mean.
// ---------------------------------------------------------------------------
__global__ __launch_bounds__(BLK)
void finalize_kernel(const float* __restrict__ partials, float* __restrict__ out) {
  float s = 0.0f;
  for (int i = threadIdx.x; i < NBLK; i += BLK) s += partials[i];
  float tot = block_sum(s);
  if (threadIdx.x == 0) out[0] = tot / (float)NTOT;
}

// ---------------------------------------------------------------------------
extern "C" void kernel_launch(void* const* d_in, const int* in_sizes, int n_in,
                              void* d_out, int out_size, void* d_ws, size_t ws_size,
                              hipStream_t stream) {
  const float* in1  = (const float*)d_in[0];   // input1  [1,3,1080,1920]
  const float* prev = (const float*)d_in[1];   // prev1   [1,3,1080,1920]
  const float* flow = (const float*)d_in[2];   // flow1   [1,2,1080,1920]
  const float* m0   = (const float*)d_in[3];   // mask1_0 [1,3,1080,1920]
  const float* ex   = (const float*)d_in[4];   // exclusive_mask1
  const int*   flag = (const int*)d_in[5];     // no_warping (scalar)

  // scratch layout: [occ u8: NPIX bytes][pad to 256][partials: NBLK floats]
  unsigned char* occ = (unsigned char*)d_ws;
  float* partials = (float*)((char*)d_ws + (((size_t)NPIX + 255) & ~(size_t)255));

  occ_kernel<<<NBLK, BLK, 0, stream>>>(flow, flag, occ);
  loss_kernel<<<NBLK, BLK, 0, stream>>>(in1, prev, flow, m0, ex, flag, occ, partials);
  finalize_kernel<<<1, BLK, 0, stream>>>(partials, (float*)d_out);
}